// BERT_CRF_22393959481361
// MI455X (gfx1250) — compile-verified
//
#include <hip/hip_runtime.h>
#include <hip/hip_bf16.h>

// ---------------------------------------------------------------------------
// BERT-CRF forward score on MI455X (gfx1250, wave32).
//
//  K1: feats = hidden(32768x768) @ W^T(768x12) + b   via V_WMMA_F32_16X16X4_F32
//      (HBM-bound: ~100MB @ 23.3TB/s => ~4.3us floor; fp32 WMMA keeps compute free)
//  K2: log-semiring associative rewrite of the 32768-step scan:
//      per-wave 32-step transfer-matrix composition (lanes 0-11 = low half,
//      lanes 16-27 = high half of a 64-step chunk), LDS fold 8 chunks/sequence.
//  K3: gold score gathers (per-sequence block reduction)
//  K4: single-wave finisher: 64 seq matrix-vector chain + logsumexp + cumsum.
// ---------------------------------------------------------------------------

typedef float v2f __attribute__((ext_vector_type(2)));
typedef float v8f __attribute__((ext_vector_type(8)));

#define BB   64
#define SS   512
#define HH   768
#define TT   12
#define START_TAG 10
#define STOP_TAG  11
#define NEGV (-10000.0f)

// ---------------- Kernel 1: feats GEMM via fp32 WMMA -----------------------
// One wave per 16-row M tile (2048 tiles). N padded 12->16 with zero rows of W
// staged in LDS. A fragment (16x4 f32, 2 VGPRs): lanes 0-15 hold K=k0,k0+1 of
// row M=lane; lanes 16-31 hold K=k0+2,k0+3. B fragment mirrors with N=lane&15.
__global__ __launch_bounds__(256) void feats_wmma_kernel(
    const float* __restrict__ hidden, const float* __restrict__ W,
    const float* __restrict__ bias, float* __restrict__ feats) {
  __shared__ float Wlds[16 * HH];
  __shared__ float blds[16];
  for (int i = threadIdx.x; i < 16 * HH; i += 256) {
    int n = i / HH, k = i % HH;
    Wlds[i] = (n < TT) ? W[n * HH + k] : 0.0f;
  }
  if (threadIdx.x < 16)
    blds[threadIdx.x] = (threadIdx.x < TT) ? bias[threadIdx.x] : 0.0f;
  __syncthreads();

  const int wave = threadIdx.x >> 5;
  const int lane = threadIdx.x & 31;
  const int tile = blockIdx.x * 8 + wave;          // 2048 tiles total
  const long row_base = (long)tile * 16;
  const int half = lane >> 4;                      // 0: K+0..1, 1: K+2..3
  const int mrow = lane & 15;
  const int ncol = lane & 15;

  const float* arow = hidden + (row_base + mrow) * HH + half * 2;
  const float* brow = &Wlds[ncol * HH + half * 2];

  v8f acc0 = {}; v8f acc1 = {};
  #pragma unroll 4
  for (int k0 = 0; k0 < HH; k0 += 8) {
    v2f a0 = *(const v2f*)(arow + k0);
    v2f b0 = *(const v2f*)(brow + k0);
    acc0 = __builtin_amdgcn_wmma_f32_16x16x4_f32(false, a0, false, b0,
                                                 (short)0, acc0, false, false);
    v2f a1 = *(const v2f*)(arow + k0 + 4);
    v2f b1 = *(const v2f*)(brow + k0 + 4);
    acc1 = __builtin_amdgcn_wmma_f32_16x16x4_f32(false, a1, false, b1,
                                                 (short)0, acc1, false, false);
  }
  v8f c = acc0 + acc1;

  // C layout: VGPR r -> (M = half*8 + r, N = lane&15)
  const float bv = blds[ncol];
  if (ncol < TT) {
    const int mb = half * 8;
    #pragma unroll
    for (int r = 0; r < 8; ++r)
      feats[(row_base + mb + r) * TT + ncol] = c[r] + bv;
  }
}

// ------------- log-semiring helper: rn[t] = LSE_j(M[t][j] + r[j]) ----------
// (inlined manually below to keep everything statically indexed)

// ---------------- Kernel 2: chunk transfer matrices + per-seq fold ---------
// Block = 8 waves = one sequence. Wave w owns 64-step chunk w; lanes 0-11
// compose steps [0,32), lanes 16-27 compose steps [32,64) (lane = column of
// the running 12x12 matrix -> zero cross-lane traffic). Then chunk = Rhi⊗Rlo,
// then wave 0 folds the 8 chunk matrices into the sequence matrix.
__global__ __launch_bounds__(256) void crf_chunks_kernel(
    const float* __restrict__ feats, const float* __restrict__ trans,
    float* __restrict__ Mseq) {
  __shared__ float lds_high[8][144];
  __shared__ float lds_chunk[8][144];

  const int wave = threadIdx.x >> 5;
  const int lane = threadIdx.x & 31;
  const int b = blockIdx.x;
  const int half = (lane >= 16) ? 1 : 0;
  int col = lane & 15; if (col > 11) col = 11;     // lanes 12-15/28-31 duplicate

  float tr[144];                                    // transitions, per-lane regs
  #pragma unroll
  for (int i = 0; i < 144; ++i) tr[i] = trans[i];

  const long base = (long)b * SS + wave * 64 + half * 32;
  const float* fp = feats + base * TT;

  float r[12], rn[12], fl[12];
  #pragma unroll
  for (int t = 0; t < TT; ++t) r[t] = tr[t * 12 + col] + fp[t];

  #pragma unroll 1
  for (int i = 1; i < 32; ++i) {                    // serial depth 32
    const float* f = fp + i * TT;
    #pragma unroll
    for (int t = 0; t < TT; ++t) fl[t] = f[t];
    #pragma unroll
    for (int t = 0; t < TT; ++t) {
      float m = -3.4e38f;
      #pragma unroll
      for (int j = 0; j < TT; ++j) m = fmaxf(m, tr[t * 12 + j] + r[j]);
      float s = 0.0f;
      #pragma unroll
      for (int j = 0; j < TT; ++j) s += __expf(tr[t * 12 + j] + r[j] - m);
      rn[t] = fl[t] + m + __logf(s);
    }
    #pragma unroll
    for (int t = 0; t < TT; ++t) r[t] = rn[t];
  }

  if (half && (lane - 16) < TT) {                   // publish Rhigh
    #pragma unroll
    for (int t = 0; t < TT; ++t) lds_high[wave][t * 12 + (lane - 16)] = r[t];
  }
  __syncthreads();

  if (!half && lane < TT) {                         // chunk = Rhigh ⊗ Rlow
    #pragma unroll
    for (int t = 0; t < TT; ++t) {
      float m = -3.4e38f;
      #pragma unroll
      for (int j = 0; j < TT; ++j) m = fmaxf(m, lds_high[wave][t * 12 + j] + r[j]);
      float s = 0.0f;
      #pragma unroll
      for (int j = 0; j < TT; ++j) s += __expf(lds_high[wave][t * 12 + j] + r[j] - m);
      rn[t] = m + __logf(s);
    }
    #pragma unroll
    for (int t = 0; t < TT; ++t) lds_chunk[wave][t * 12 + lane] = rn[t];
  }
  __syncthreads();

  if (wave == 0 && lane < TT) {                     // fold 8 chunks (depth 7)
    #pragma unroll
    for (int t = 0; t < TT; ++t) r[t] = lds_chunk[0][t * 12 + lane];
    #pragma unroll 1
    for (int n = 1; n < 8; ++n) {
      #pragma unroll
      for (int t = 0; t < TT; ++t) {
        float m = -3.4e38f;
        #pragma unroll
        for (int j = 0; j < TT; ++j) m = fmaxf(m, lds_chunk[n][t * 12 + j] + r[j]);
        float s = 0.0f;
        #pragma unroll
        for (int j = 0; j < TT; ++j) s += __expf(lds_chunk[n][t * 12 + j] + r[j] - m);
        rn[t] = m + __logf(s);
      }
      #pragma unroll
      for (int t = 0; t < TT; ++t) r[t] = rn[t];
    }
    #pragma unroll
    for (int t = 0; t < TT; ++t) Mseq[b * 144 + t * 12 + lane] = r[t];
  }
}

// ---------------- Kernel 3: gold path (per-sequence reduction) -------------
__global__ __launch_bounds__(256) void gold_kernel(
    const float* __restrict__ feats, const float* __restrict__ trans,
    const int* __restrict__ tags, float* __restrict__ gpart) {
  __shared__ float red[256];
  const int b = blockIdx.x;
  const int t = threadIdx.x;
  float sum = 0.0f;
  for (int s = t; s < SS; s += 256) {
    const int nxt = tags[b * SS + s];
    const int prv = (s == 0) ? START_TAG : tags[b * SS + s - 1];
    sum += trans[nxt * TT + prv];
    sum += feats[((long)b * SS + s) * TT + nxt];
  }
  red[t] = sum;
  __syncthreads();
  for (int off = 128; off > 0; off >>= 1) {
    if (t < off) red[t] += red[t + off];
    __syncthreads();
  }
  if (t == 0) gpart[b] = red[0] + trans[STOP_TAG * TT + tags[b * SS + SS - 1]];
}

// ---------------- Kernel 4: single-wave finisher ---------------------------
__global__ __launch_bounds__(32) void final_kernel(
    const float* __restrict__ trans, const float* __restrict__ Mseq,
    const float* __restrict__ gpart, float* __restrict__ out) {
  __shared__ float fv[16];
  __shared__ float term[16];
  __shared__ float fsArr[BB];
  const int lane = threadIdx.x;
  if (lane < TT) fv[lane] = (lane == START_TAG) ? 0.0f : NEGV;
  __syncthreads();

  for (int b = 0; b < BB; ++b) {
    float nf = 0.0f;
    if (lane < TT) {                                // fv' = Mseq_b ⊗ fv
      const float* M = Mseq + b * 144 + lane * 12;  // row 'lane'
      float m = -3.4e38f;
      #pragma unroll
      for (int j = 0; j < TT; ++j) m = fmaxf(m, M[j] + fv[j]);
      float s = 0.0f;
      #pragma unroll
      for (int j = 0; j < TT; ++j) s += __expf(M[j] + fv[j] - m);
      nf = m + __logf(s);
    }
    __syncthreads();
    if (lane < TT) {
      fv[lane] = nf;
      term[lane] = nf + trans[STOP_TAG * TT + lane];
    }
    __syncthreads();
    if (lane == 0) {                                // forward_score[b]
      float m = term[0];
      for (int j = 1; j < TT; ++j) m = fmaxf(m, term[j]);
      float s = 0.0f;
      for (int j = 0; j < TT; ++j) s += __expf(term[j] - m);
      fsArr[b] = m + __logf(s);
    }
    __syncthreads();
  }

  if (lane == 0) {
    float cum = 0.0f, tot = 0.0f;
    for (int b = 0; b < BB; ++b) { cum += gpart[b]; tot += fsArr[b] - cum; }
    out[0] = tot;
  }
}

// ---------------------------------------------------------------------------
extern "C" void kernel_launch(void* const* d_in, const int* in_sizes, int n_in,
                              void* d_out, int out_size, void* d_ws, size_t ws_size,
                              hipStream_t stream) {
  const float* hidden = (const float*)d_in[0];   // (64,512,768) f32
  const float* W      = (const float*)d_in[1];   // (12,768)     f32
  const float* bias   = (const float*)d_in[2];   // (12,)        f32
  const float* trans  = (const float*)d_in[3];   // (12,12)      f32
  const int*   tags   = (const int*)d_in[4];     // (64,512)     int
  float* out = (float*)d_out;

  float* feats = (float*)d_ws;                   // 32768*12 = 393216 floats
  float* Mseq  = feats + 393216;                 // 64*144   =   9216 floats
  float* gpart = Mseq + 9216;                    // 64 floats

  feats_wmma_kernel<<<256, 256, 0, stream>>>(hidden, W, bias, feats);
  crf_chunks_kernel<<<64, 256, 0, stream>>>(feats, trans, Mseq);
  gold_kernel<<<64, 256, 0, stream>>>(feats, trans, tags, gpart);
  final_kernel<<<1, 32, 0, stream>>>(trans, Mseq, gpart, out);
}